// BipartiteNeuralMessagePassingLayer_73237782331672
// MI455X (gfx1250) — compile-verified
//
#include <hip/hip_runtime.h>
#include <limits>

typedef float v2f __attribute__((ext_vector_type(2)));
typedef float v8f __attribute__((ext_vector_type(8)));

#define A_ROWS 1024
#define B_ROWS 1024
#define EDIM   16     // EDGE_DIM
#define NDIM   32     // NODE_DIM
#define EIN    32     // 2*EDGE_DIM = edge input features
#define AITER  8      // a-rows per wave in main kernel
#define WPB    8      // waves per block (256 threads, wave32)

// single-instruction ReLU: v_med3_f32(x, 0, +inf) == clamp(x, 0, inf)
__device__ __forceinline__ float relu1(float x) {
    return __builtin_amdgcn_fmed3f(x, 0.0f, __builtin_inff());
}

// ---------------------------------------------------------------------------
// Kernel 1: an_proj = nodes_a@Wa + b1e ; bn_proj = nodes_b@Wb ; zero the sums
// ---------------------------------------------------------------------------
__global__ void prep_kernel(const float* __restrict__ nodes_a,
                            const float* __restrict__ nodes_b,
                            const float* __restrict__ W1e,   // (96,16)
                            const float* __restrict__ b1e,   // (16)
                            float* __restrict__ an_proj,     // (1024,16)
                            float* __restrict__ bn_proj,     // (1024,16)
                            float* __restrict__ a_sum,       // (1024,16)
                            float* __restrict__ b_sum)       // (1024,16)
{
    int t   = blockIdx.x * blockDim.x + threadIdx.x;   // 0..2047
    int row = t & 1023;
    bool isB = t >= 1024;
    const float* x = (isB ? nodes_b : nodes_a) + row * NDIM;
    const float* W = W1e + (isB ? 64 : 32) * EDIM;     // Wa rows 32..63, Wb rows 64..95
    float acc[EDIM];
    #pragma unroll
    for (int d = 0; d < EDIM; ++d) acc[d] = isB ? 0.f : b1e[d];
    for (int k = 0; k < NDIM; ++k) {
        float xv = x[k];
        const float* wr = W + k * EDIM;
        #pragma unroll
        for (int d = 0; d < EDIM; ++d) acc[d] = fmaf(xv, wr[d], acc[d]);
    }
    float* dst = (isB ? bn_proj : an_proj) + row * EDIM;
    float* zs  = (isB ? b_sum   : a_sum  ) + row * EDIM;
    #pragma unroll
    for (int d = 0; d < EDIM; ++d) { dst[d] = acc[d]; zs[d] = 0.f; }
}

// ---------------------------------------------------------------------------
// Kernel 2: edge MLP via fp32 WMMA. One wave = 16-b strip, loops over 8 a's.
// ---------------------------------------------------------------------------
__global__ __launch_bounds__(256) void edge_kernel(
    const float* __restrict__ edge_embeds,   // (1024,1024,32)
    const float* __restrict__ W1e,           // rows 0..31 = We (32,16)
    const float* __restrict__ W2e,           // (16,16)
    const float* __restrict__ b2e,           // (16)
    const float* __restrict__ an_proj,       // (1024,16) includes b1e
    const float* __restrict__ bn_proj,       // (1024,16)
    float* __restrict__ edge_latent,         // (1024,1024,16)  -> d_out
    float* __restrict__ a_sum,               // (1024,16)
    float* __restrict__ b_sum)               // (1024,16)
{
    __shared__ float lds[WPB][16 * 18];      // stride 18: bank-conflict-free, 8B aligned
    const int lane = threadIdx.x & 31;
    const int wav  = threadIdx.x >> 5;
    const int g    = blockIdx.x * WPB + wav;
    const int b0   = (g & 63) * 16;          // 64 strips of 16 b's
    const int a0   = (g >> 6) * AITER;       // 128 a-groups of 8

    const int n    = lane & 15;              // N / feature index for B,C,D frags
    const int half = lane >> 4;              // lanes 16..31 hold K+2/K+3 (A,B), M+8 (C,D)
    const int koff = half * 2;
    const int m    = n;                      // A-matrix row index for this lane

    // Preload We B-fragments: chunk kk covers K = 4*kk .. 4*kk+3
    v2f bf1[8];
    #pragma unroll
    for (int kk = 0; kk < 8; ++kk) {
        bf1[kk].x = W1e[(kk * 4 + koff + 0) * EDIM + n];
        bf1[kk].y = W1e[(kk * 4 + koff + 1) * EDIM + n];
    }
    // Preload W2e B-fragments (K = 16 -> 4 chunks)
    v2f bf2[4];
    #pragma unroll
    for (int c = 0; c < 4; ++c) {
        bf2[c].x = W2e[(c * 4 + koff + 0) * EDIM + n];
        bf2[c].y = W2e[(c * 4 + koff + 1) * EDIM + n];
    }
    // bn_proj rows of this strip, pre-arranged in C/D layout
    float bn[8];
    #pragma unroll
    for (int r = 0; r < 8; ++r)
        bn[r] = bn_proj[(b0 + r + half * 8) * EDIM + n];
    const float b2 = b2e[n];

    float bacc[8];                           // b_sum accumulator (register resident)
    #pragma unroll
    for (int r = 0; r < 8; ++r) bacc[r] = 0.f;

    float* myLds = lds[wav];

    for (int i = 0; i < AITER; ++i) {
        const int a = a0 + i;
        const float an = an_proj[a * EDIM + n];

        const float* xrow = edge_embeds + ((size_t)a * B_ROWS + b0 + m) * (size_t)EIN;
        if (i + 1 < AITER)   // global_prefetch_b8 for next strip
            __builtin_prefetch(edge_embeds + ((size_t)(a + 1) * B_ROWS + b0 + m) * (size_t)EIN, 0, 0);

        // Layer 1: h = X @ We + an + bn  (C seeded with node projections + bias)
        v8f c1;
        #pragma unroll
        for (int r = 0; r < 8; ++r) c1[r] = an + bn[r];
        #pragma unroll
        for (int kk = 0; kk < 8; ++kk) {
            // stream-once data: non-temporal aligned global_load_b64
            v2f af = __builtin_nontemporal_load((const v2f*)(xrow + kk * 4 + koff));
            c1 = __builtin_amdgcn_wmma_f32_16x16x4_f32(
                     false, af, false, bf1[kk], (short)0, c1, false, false);
        }
        // ReLU (single v_med3_f32), stage to LDS: D-layout -> A-layout
        #pragma unroll
        for (int r = 0; r < 8; ++r)
            myLds[(r + half * 8) * 18 + n] = relu1(c1[r]);
        __asm__ volatile("s_wait_dscnt 0" ::: "memory");    // wave-synchronous LDS RAW

        // Layer 2: y = relu(h) @ W2e + b2e
        v8f c2;
        #pragma unroll
        for (int r = 0; r < 8; ++r) c2[r] = b2;
        #pragma unroll
        for (int c = 0; c < 4; ++c) {
            v2f af = *(const v2f*)(myLds + m * 18 + c * 4 + koff);  // ds_load_b64
            c2 = __builtin_amdgcn_wmma_f32_16x16x4_f32(
                     false, af, false, bf2[c], (short)0, c2, false, false);
        }

        float s = 0.f;
        float* orow = edge_latent + ((size_t)a * B_ROWS + b0) * (size_t)EDIM;
        #pragma unroll
        for (int r = 0; r < 8; ++r) {
            float y = relu1(c2[r]);
            // write-once output, never re-read on chip: non-temporal store
            __builtin_nontemporal_store(y, orow + (r + half * 8) * EDIM + n);
            bacc[r] += y;                          // b_sum partial, in registers
            s += y;                                // a_sum partial over 8 rows
        }
        atomicAdd(a_sum + a * EDIM + n, s);        // lanes L and L+16 both contribute
    }
    #pragma unroll
    for (int r = 0; r < 8; ++r)                    // one b_sum flush per wave
        atomicAdd(b_sum + (b0 + r + half * 8) * EDIM + n, bacc[r]);
}

// ---------------------------------------------------------------------------
// Kernel 3: node MLPs on concat(nodes, edge_latent sums). 2048 rows, tiny.
// ---------------------------------------------------------------------------
__global__ void node_kernel(const float* __restrict__ nodes_a,
                            const float* __restrict__ nodes_b,
                            const float* __restrict__ a_sum,
                            const float* __restrict__ b_sum,
                            const float* __restrict__ W1n,   // (48,32)
                            const float* __restrict__ b1n,   // (32)
                            const float* __restrict__ W2n,   // (32,32)
                            const float* __restrict__ b2n,   // (32)
                            float* __restrict__ out_a,       // (1024,32)
                            float* __restrict__ out_b)       // (1024,32)
{
    int t   = blockIdx.x * blockDim.x + threadIdx.x;   // 0..2047
    int row = t & 1023;
    bool isB = t >= 1024;
    const float* nx = (isB ? nodes_b : nodes_a) + row * NDIM;
    const float* sx = (isB ? b_sum   : a_sum  ) + row * EDIM;
    float h[NDIM];
    #pragma unroll
    for (int j = 0; j < NDIM; ++j) h[j] = b1n[j];
    for (int k = 0; k < NDIM; ++k) {
        float xv = nx[k];
        #pragma unroll
        for (int j = 0; j < NDIM; ++j) h[j] = fmaf(xv, W1n[k * NDIM + j], h[j]);
    }
    for (int k = 0; k < EDIM; ++k) {
        float xv = sx[k];
        #pragma unroll
        for (int j = 0; j < NDIM; ++j) h[j] = fmaf(xv, W1n[(NDIM + k) * NDIM + j], h[j]);
    }
    #pragma unroll
    for (int j = 0; j < NDIM; ++j) h[j] = relu1(h[j]);
    float o[NDIM];
    #pragma unroll
    for (int j = 0; j < NDIM; ++j) o[j] = b2n[j];
    for (int k = 0; k < NDIM; ++k) {
        float xv = h[k];
        #pragma unroll
        for (int j = 0; j < NDIM; ++j) o[j] = fmaf(xv, W2n[k * NDIM + j], o[j]);
    }
    float* dst = (isB ? out_b : out_a) + row * NDIM;
    #pragma unroll
    for (int j = 0; j < NDIM; ++j) dst[j] = relu1(o[j]);
}

// ---------------------------------------------------------------------------
extern "C" void kernel_launch(void* const* d_in, const int* in_sizes, int n_in,
                              void* d_out, int out_size, void* d_ws, size_t ws_size,
                              hipStream_t stream) {
    (void)in_sizes; (void)n_in; (void)out_size; (void)ws_size;
    const float* edge_embeds = (const float*)d_in[0];
    const float* nodes_a     = (const float*)d_in[1];
    const float* nodes_b     = (const float*)d_in[2];
    const float* W1e         = (const float*)d_in[3];
    const float* b1e         = (const float*)d_in[4];
    const float* W2e         = (const float*)d_in[5];
    const float* b2e         = (const float*)d_in[6];
    const float* W1n         = (const float*)d_in[7];
    const float* b1n         = (const float*)d_in[8];
    const float* W2n         = (const float*)d_in[9];
    const float* b2n         = (const float*)d_in[10];

    float* out         = (float*)d_out;
    float* edge_latent = out;                                   // 1024*1024*16
    float* out_a       = out + (size_t)A_ROWS * B_ROWS * EDIM;  // 1024*32
    float* out_b       = out_a + A_ROWS * NDIM;

    // workspace: 4 x (1024*16) floats = 256 KB
    float* ws      = (float*)d_ws;
    float* an_proj = ws;
    float* bn_proj = ws + 16384;
    float* a_sum   = ws + 32768;
    float* b_sum   = ws + 49152;

    prep_kernel<<<8, 256, 0, stream>>>(nodes_a, nodes_b, W1e, b1e,
                                       an_proj, bn_proj, a_sum, b_sum);
    edge_kernel<<<1024, 256, 0, stream>>>(edge_embeds, W1e, W2e, b2e,
                                          an_proj, bn_proj,
                                          edge_latent, a_sum, b_sum);
    node_kernel<<<8, 256, 0, stream>>>(nodes_a, nodes_b, a_sum, b_sum,
                                       W1n, b1n, W2n, b2n, out_a, out_b);
}